// GraphFeatureTokenizer_56160992362538
// MI455X (gfx1250) — compile-verified
//
#include <hip/hip_runtime.h>
#include <hip/hip_bf16.h>

typedef __attribute__((ext_vector_type(2))) float v2f;
typedef __attribute__((ext_vector_type(8))) float v8f;

// Problem constants (from the reference)
constexpr int B_    = 4;
constexpr int N_PER = 4096;
constexpr int E_PER = 8192;
constexpr int F_    = 2;
constexpr int S_    = 4;
constexpr int K_    = 16;
constexpr int D_    = 256;
constexpr int TN_   = B_ * N_PER;   // 16384
constexpr int TE_   = B_ * E_PER;   // 32768
constexpr int T_    = 2 * E_PER;    // 16384 = 2*max(N_PER,E_PER)
constexpr int TP2_  = T_ + 2;       // after graph/null token concat

constexpr long FEAT_ELEMS = (long)B_ * TP2_ * F_ * D_;
constexpr long MASK_ELEMS = (long)B_ * TP2_;

// ---------------------------------------------------------------------------
// Kernel 1: base features (embedding sums + order emb + graph/null tokens),
// padding mask, padded_index. One thread per (b, t, d); writes both F slices.
// ---------------------------------------------------------------------------
__global__ void tok_base_kernel(
    const int*   __restrict__ node_data,   // (TN,F,S)
    const int*   __restrict__ edge_data,   // (TE,F,S)
    const int*   __restrict__ edge_index,  // (2,TE)
    const int*   __restrict__ node_num,    // (B,)
    const int*   __restrict__ edge_num,    // (B,)
    const float* __restrict__ atom_emb,    // (VOCAB,D)
    const float* __restrict__ edge_emb,    // (VOCAB,D)
    const float* __restrict__ order_emb,   // (2,D)
    const float* __restrict__ graph_tok,   // (1,D)
    const float* __restrict__ null_tok,    // (1,D)
    float*       __restrict__ out)         // feat | mask | index (flat)
{
    long tid = (long)blockIdx.x * blockDim.x + threadIdx.x;
    long total = (long)B_ * TP2_ * D_;
    if (tid >= total) return;

    int d = (int)(tid % D_);
    int t = (int)((tid / D_) % TP2_);
    int b = (int)(tid / ((long)D_ * TP2_));

    float* feat  = out;
    float* masko = out + FEAT_ELEMS;
    float* idxo  = masko + MASK_ELEMS;

    long fbase = (((long)b * TP2_ + t) * F_) * D_ + d;

    if (t == 0) {
        float v = graph_tok[d];
        feat[fbase] = v; feat[fbase + D_] = v;
        if (d == 0) masko[(long)b * TP2_ + t] = 0.0f;
        return;
    }
    if (t == 1) {
        float v = null_tok[d];
        feat[fbase] = v; feat[fbase + D_] = v;
        if (d == 0) masko[(long)b * TP2_ + t] = 0.0f;
        return;
    }

    int tp = t - 2;
    int nn = node_num[b];
    int en = edge_num[b];
    int seq = nn > en ? nn : en;

    int noffb = 0, eoffb = 0;
    for (int i = 0; i < b; ++i) { noffb += node_num[i]; eoffb += edge_num[i]; }

    bool isNode = (tp < nn);
    bool isEdge = (!isNode) && (tp < nn + en);
    int idx0 = 0, idx1 = 0;
    if (isNode) { idx0 = tp; idx1 = tp; }
    else if (isEdge) {
        int e = eoffb + (tp - nn);
        idx0 = edge_index[e];
        idx1 = edge_index[TE_ + e];
    }

    bool masked = (tp >= seq);

    if (d == 0) {
        idxo[((long)b * T_ + tp) * 2 + 0] = (float)idx0;
        idxo[((long)b * T_ + tp) * 2 + 1] = (float)idx1;
        masko[(long)b * TP2_ + t] = masked ? 1.0f : 0.0f;
    }

    if (masked) {
        feat[fbase] = 0.0f; feat[fbase + D_] = 0.0f;
        return;
    }

    int order = (idx0 == idx1) ? 1 : 0;
    float oe = order_emb[order * D_ + d];

    float v0, v1;
    if (isNode) {
        long row = (long)(noffb + tp);
        const int* nd = node_data + row * (F_ * S_);
        v0 = 0.0f; v1 = 0.0f;
        #pragma unroll
        for (int s = 0; s < S_; ++s) {
            v0 += atom_emb[(long)nd[s]       * D_ + d];
            v1 += atom_emb[(long)nd[S_ + s]  * D_ + d];
        }
    } else if (isEdge) {
        long e = (long)eoffb + (tp - nn);
        const int* ed = edge_data + e * (F_ * S_);
        v0 = 0.0f; v1 = 0.0f;
        #pragma unroll
        for (int s = 0; s < S_; ++s) {
            v0 += edge_emb[(long)ed[s]       * D_ + d];
            v1 += edge_emb[(long)ed[S_ + s]  * D_ + d];
        }
    } else {
        v0 = 1.0f; v1 = 1.0f;   // padded_feature initialized to ones
    }

    feat[fbase]      = v0 + oe;
    feat[fbase + D_] = v1 + oe;
}

// ---------------------------------------------------------------------------
// Kernel 2: lap-eigvec projection via WMMA f32 16x16x4 chains.
//   lapterm[token, d] = sum_{j<32} idx_embed[token, j] * lap_W[d, j]
// One wave (32 lanes) per 16-token tile; 16 N-tiles of 16 columns (D=256),
// 8 chained K=4 WMMAs per tile (K=32). Accumulated into both F slices of the
// feature tensor. Fast unconditional-store path for fully-active tiles
// (the common case: seq is a multiple of 16), predicated path otherwise.
// ---------------------------------------------------------------------------
__global__ void lap_wmma_kernel(
    const int*   __restrict__ edge_index,  // (2,TE)
    const int*   __restrict__ node_num,
    const int*   __restrict__ edge_num,
    const float* __restrict__ eig,         // (TN,K)
    const float* __restrict__ lap_W,       // (D,2K) row-major
    float*       __restrict__ out)         // feature region of d_out
{
    const int wavesPerBlk = blockDim.x >> 5;
    const int gw   = blockIdx.x * wavesPerBlk + (threadIdx.x >> 5);
    const int lane = threadIdx.x & 31;
    const int tilesPerBatch = T_ / 16;     // 1024

    int b    = gw / tilesPerBatch;
    int tile = gw % tilesPerBatch;
    if (b >= B_) return;                   // wave-uniform

    int nn = node_num[b];
    int en = edge_num[b];
    int seq = nn > en ? nn : en;
    int tp0 = tile * 16;
    if (tp0 >= seq) return;                // wave-uniform: masked tile, adds nothing

    const bool fullTile = (tp0 + 16 <= seq);   // wave-uniform

    int noffb = 0, eoffb = 0;
    for (int i = 0; i < b; ++i) { noffb += node_num[i]; eoffb += edge_num[i]; }

    const int m  = lane & 15;              // M row within tile / N col within tile
    const int hi = lane >> 4;              // K-pair selector per 16x4 f32 A layout
    int tp = tp0 + m;

    // Per-token eigvec row indices (idx0/idx1); inactive rows read row 0 (safe).
    int idx0 = 0, idx1 = 0;
    if (tp < seq) {
        if (tp < nn) { idx0 = tp; idx1 = tp; }
        else if (tp < nn + en) {
            int e = eoffb + (tp - nn);
            idx0 = edge_index[e];
            idx1 = edge_index[TE_ + e];
        }
    }
    const float* r0 = eig + (long)(noffb + idx0) * K_;
    const float* r1 = eig + (long)(noffb + idx1) * K_;

    // A fragments: slice kk covers K = 4kk..4kk+3.
    // Lane (m,hi): VGPR0 holds K=4kk+2hi, VGPR1 holds K=4kk+2hi+1.
    // j < 16 -> first eigvec row, j >= 16 -> second.
    v2f a[8];
    #pragma unroll
    for (int kk = 0; kk < 8; ++kk) {
        int j = kk * 4 + hi * 2;
        if (kk < 4) { a[kk][0] = r0[j];      a[kk][1] = r0[j + 1];      }
        else        { a[kk][0] = r1[j - 16]; a[kk][1] = r1[j - 15];     }
    }

    // Base offset for this lane's output rows: row M = 8*hi is the first row
    // this lane stores (C/D layout: VGPR r -> M = r + 8*hi, N = lane&15).
    const long rowStride = (long)F_ * D_;  // floats between consecutive tokens

    // Loop over 16 column tiles of D=256.
    for (int nt = 0; nt < 16; ++nt) {
        int dcol = nt * 16 + m;            // this lane's N column
        const float* wrow = lap_W + (long)dcol * (2 * K_);  // lap_W[dcol][j]

        v8f acc = {0.f, 0.f, 0.f, 0.f, 0.f, 0.f, 0.f, 0.f};
        #pragma unroll
        for (int kk = 0; kk < 8; ++kk) {
            int j = kk * 4 + hi * 2;
            v2f bm;
            bm[0] = wrow[j];
            bm[1] = wrow[j + 1];
            // D = A(16x4,f32) * B(4x16,f32) + C
            acc = __builtin_amdgcn_wmma_f32_16x16x4_f32(
                false, a[kk], false, bm, (short)0, acc, false, false);
        }

        long fbase = (((long)b * TP2_ + (tp0 + 2 + 8 * hi)) * F_) * D_ + dcol;

        if (fullTile) {
            // Unconditional RMW: 8 rows x 2 F-slices, immediate-offset friendly.
            #pragma unroll
            for (int r = 0; r < 8; ++r) {
                long fb = fbase + (long)r * rowStride;
                float x0 = out[fb];
                float x1 = out[fb + D_];
                out[fb]      = x0 + acc[r];
                out[fb + D_] = x1 + acc[r];
            }
        } else {
            // Generic partial tile: predicate per row.
            #pragma unroll
            for (int r = 0; r < 8; ++r) {
                int tpr = tp0 + r + 8 * hi;
                if (tpr < seq) {
                    long fb = fbase + (long)r * rowStride;
                    out[fb]      += acc[r];
                    out[fb + D_] += acc[r];
                }
            }
        }
    }
}

// ---------------------------------------------------------------------------
extern "C" void kernel_launch(void* const* d_in, const int* in_sizes, int n_in,
                              void* d_out, int out_size, void* d_ws, size_t ws_size,
                              hipStream_t stream) {
    (void)in_sizes; (void)n_in; (void)out_size; (void)d_ws; (void)ws_size;

    const int*   node_data  = (const int*)  d_in[0];
    const int*   edge_data  = (const int*)  d_in[1];
    const int*   edge_index = (const int*)  d_in[2];
    const int*   node_num   = (const int*)  d_in[3];
    const int*   edge_num   = (const int*)  d_in[4];
    const float* lap_eigvec = (const float*)d_in[5];
    const float* atom_emb   = (const float*)d_in[6];
    const float* edge_emb   = (const float*)d_in[7];
    const float* lap_W      = (const float*)d_in[8];
    const float* order_emb  = (const float*)d_in[9];
    const float* graph_tok  = (const float*)d_in[10];
    const float* null_tok   = (const float*)d_in[11];
    float* out = (float*)d_out;

    // Kernel 1: base features + mask + index. One thread per (b,t,d).
    {
        long total = (long)B_ * TP2_ * D_;            // 16,779,264
        int threads = 256;
        int blocks = (int)((total + threads - 1) / threads);
        tok_base_kernel<<<blocks, threads, 0, stream>>>(
            node_data, edge_data, edge_index, node_num, edge_num,
            atom_emb, edge_emb, order_emb, graph_tok, null_tok, out);
    }

    // Kernel 2: WMMA lap projection accumulate. 1 wave per 16-token tile.
    {
        int totalWaves = B_ * (T_ / 16);              // 4096 waves
        int threads = 256;                            // 8 waves/block
        int blocks = totalWaves / (threads / 32);     // 512 blocks
        lap_wmma_kernel<<<blocks, threads, 0, stream>>>(
            edge_index, node_num, edge_num, lap_eigvec, lap_W, out);
    }
}